// VectorQuant_64862596104495
// MI455X (gfx1250) — compile-verified
//
#include <hip/hip_runtime.h>

typedef float v2f __attribute__((ext_vector_type(2)));
typedef float v8f __attribute__((ext_vector_type(8)));

// Problem constants (from reference): N=8, S=2048, C=4, K=512, V=64
constexpr int C_ = 4;
constexpr int K_ = 512;
constexpr int V_ = 64;
constexpr int ROWS = 8 * 2048;          // N*S = 16384
constexpr int RT = 2;                   // row-tiles (of 16 rows) per wave
constexpr int WAVES = 8;                // waves per block
constexpr int ROWS_PER_BLOCK = WAVES * RT * 16;  // 256
constexpr int NT = K_ / 16;             // 32 codebook tiles of 16
// LDS: B staged in WMMA-fragment order: (tile,step,lane) -> float2
//   entry e = ((tau*16 + t)*32 + L), value = emb[c][tau*16 + L%16][4t + 2*(L/16) .. +1]
constexpr int LDS_B_F2 = NT * 16 * 32;          // 16384 float2 = 128 KB
constexpr int LDS_FLOATS = LDS_B_F2 * 2 + K_;   // + e2[512]

__global__ __launch_bounds__(256)
void vq_wmma_kernel(const float* __restrict__ x, const float* __restrict__ emb,
                    float* __restrict__ out0, float* __restrict__ out1,
                    float* __restrict__ out2)
{
    extern __shared__ float smem[];
    v2f*   ldsB  = (v2f*)smem;                 // swizzled B fragments
    float* ldsE2 = smem + LDS_B_F2 * 2;        // e2 per codeword

    const int c   = blockIdx.y;
    const int tid = threadIdx.x;
    const int lane = tid & 31;
    const int wid  = tid >> 5;
    const int mrow = lane & 15;                // M (for A) or N (for B/C) sub-index
    const int hi   = lane >> 4;                // which K/M half this lane holds

    const float* __restrict__ embC = emb + (size_t)c * K_ * V_;

    // ---- Stage B into LDS in exact WMMA fragment order (coalesced LDS writes) ----
    #pragma unroll
    for (int j = 0; j < LDS_B_F2 / 256; ++j) {
        int e   = tid + 256 * j;               // 0..16383
        int L   = e & 31;
        int s   = e >> 5;                      // (tau*16 + t)
        int t   = s & 15;
        int tau = s >> 4;
        int k   = tau * 16 + (L & 15);
        int v   = 4 * t + 2 * (L >> 4);
        ldsB[e] = *(const v2f*)(embC + k * V_ + v);
    }
    // ---- e2[k] = sum_v emb[c][k][v]^2 ----
    for (int k = tid; k < K_; k += 256) {
        const float4* p = (const float4*)(embC + k * V_);
        float s = 0.f;
        #pragma unroll
        for (int i = 0; i < V_ / 4; ++i) {
            float4 q = p[i];
            s += q.x * q.x + q.y * q.y + q.z * q.z + q.w * q.w;
        }
        ldsE2[k] = s;
    }
    __syncthreads();

    // ---- Load A fragments (x rows) into registers: ISA 16x4 f32 A layout ----
    const int rowBase = blockIdx.x * ROWS_PER_BLOCK + wid * (RT * 16);
    v2f a[RT][16];
    #pragma unroll
    for (int rt = 0; rt < RT; ++rt) {
        const float* xr = x + ((size_t)(rowBase + rt * 16 + mrow) * C_ + c) * V_;
        #pragma unroll
        for (int t = 0; t < 16; ++t)
            a[rt][t] = *(const v2f*)(xr + 4 * t + 2 * hi);
    }

    // ---- GEMM + running argmin over all 32 codebook tiles ----
    float minv[RT][8];
    int   mink[RT][8];
    #pragma unroll
    for (int rt = 0; rt < RT; ++rt)
        #pragma unroll
        for (int r = 0; r < 8; ++r) { minv[rt][r] = 3.402823466e38f; mink[rt][r] = 0; }

    for (int tau = 0; tau < NT; ++tau) {
        const float e2v = ldsE2[tau * 16 + mrow];      // this lane's N = tau*16 + mrow
        const v2f* bp = ldsB + tau * 16 * 32 + lane;   // step stride = 32 float2

        v8f acc[RT];
        #pragma unroll
        for (int rt = 0; rt < RT; ++rt) { v8f z = {}; acc[rt] = z; }

        #pragma unroll
        for (int t = 0; t < 16; ++t) {
            v2f b = bp[t * 32];
            #pragma unroll
            for (int rt = 0; rt < RT; ++rt)
                acc[rt] = __builtin_amdgcn_wmma_f32_16x16x4_f32(
                    false, a[rt][t], false, b, (short)0, acc[rt], false, false);
        }
        const int kk = tau * 16 + mrow;
        #pragma unroll
        for (int rt = 0; rt < RT; ++rt)
            #pragma unroll
            for (int r = 0; r < 8; ++r) {
                float d = e2v - 2.0f * acc[rt][r];     // x2 is constant per row: drop it
                if (d < minv[rt][r]) { minv[rt][r] = d; mink[rt][r] = kk; }
            }
    }

    // ---- Cross-lane argmin within each 16-lane group (ties -> lower k) ----
    #pragma unroll
    for (int rt = 0; rt < RT; ++rt)
        #pragma unroll
        for (int r = 0; r < 8; ++r) {
            float v = minv[rt][r]; int k = mink[rt][r];
            #pragma unroll
            for (int m = 1; m <= 8; m <<= 1) {
                float v2 = __shfl_xor(v, m, 32);
                int   k2 = __shfl_xor(k, m, 32);
                if (v2 < v || (v2 == v && k2 < k)) { v = v2; k = k2; }
            }
            mink[rt][r] = k;    // uniform within each 16-lane group
        }

    // ---- Emit outputs: out0 = (e - x) + x ; out1 = out2 = sum (x - e)^2 ----
    #pragma unroll
    for (int rt = 0; rt < RT; ++rt) {
        #pragma unroll
        for (int m = 0; m < 16; ++m) {
            int kw  = __shfl(mink[rt][m & 7], (m >> 3) << 4, 32);
            int row = rowBase + rt * 16 + m;
            const v2f* ep = (const v2f*)(embC + (size_t)kw * V_);
            const v2f* xp = (const v2f*)(x + ((size_t)row * C_ + c) * V_);
            v2f e = ep[lane];
            v2f xv = xp[lane];
            v2f o;
            o.x = (e.x - xv.x) + xv.x;
            o.y = (e.y - xv.y) + xv.y;
            ((v2f*)(out0 + ((size_t)row * C_ + c) * V_))[lane] = o;
            float dx = xv.x - e.x, dy = xv.y - e.y;
            float s = dx * dx + dy * dy;
            #pragma unroll
            for (int mm = 1; mm < 32; mm <<= 1) s += __shfl_xor(s, mm, 32);
            if (lane == 0) {
                out1[(size_t)row * C_ + c] = s;
                out2[(size_t)row * C_ + c] = s;
            }
        }
    }
}

extern "C" void kernel_launch(void* const* d_in, const int* in_sizes, int n_in,
                              void* d_out, int out_size, void* d_ws, size_t ws_size,
                              hipStream_t stream) {
    const float* x   = (const float*)d_in[0];
    const float* emb = (const float*)d_in[1];
    float* out  = (float*)d_out;
    float* out0 = out;                                   // (N,S,C,V)
    float* out1 = out0 + (size_t)ROWS * C_ * V_;         // (N,S,C)
    float* out2 = out1 + (size_t)ROWS * C_;              // (N,S,C)

    dim3 grid(ROWS / ROWS_PER_BLOCK, C_);
    dim3 block(256);
    size_t smem = (size_t)LDS_FLOATS * sizeof(float);    // 130 KB < 320 KB/WGP
    hipLaunchKernelGGL(vq_wmma_kernel, grid, block, smem, stream,
                       x, emb, out0, out1, out2);
}